// AttMaxPool2D_79482664780429
// MI455X (gfx1250) — compile-verified
//
#include <hip/hip_runtime.h>

// AttMaxPool2D: x [16,224,224,128] f32 NHWC -> out [16,112,112,128] f32.
// out[b,ho,wo,c] = sum_i v_i * softmax_i(v) over each 2x2 window (stride 2, VALID).
//
// Roofline: 411 MB in + 103 MB out, each byte touched once -> ~22 us @ 23.3 TB/s.
// Pure streaming kernel; WMMA not applicable (data-dependent weights, no GEMM).
//
// Mapping (divide-free):
//   blockIdx.z = b, blockIdx.y = ho, blockIdx.x*8 + (tid>>5) = wo, lane = c4.
//   1 wave32 = 1 output pixel (128 channels), 1 lane = float4 of channels.
//   Each window load: 512 B contiguous per wave, b128 per lane, NT cache hint.

typedef float v4f __attribute__((ext_vector_type(4)));

#define B_  16
#define H_  224
#define W_  224
#define C_  128
#define HO_ 112
#define WO_ 112

__global__ __launch_bounds__(256, 8)
void attn_pool2x2_kernel(const float* __restrict__ x, float* __restrict__ out) {
    const unsigned c4 = threadIdx.x & 31u;          // lane: which float4 of C
    const unsigned wv = threadIdx.x >> 5;           // wave within block: 0..7
    const unsigned wo = blockIdx.x * 8u + wv;       // 0..111
    const unsigned ho = blockIdx.y;                 // 0..111  (scalar)
    const unsigned b  = blockIdx.z;                 // 0..15   (scalar)

    // Offsets in floats; max input offset ~103M < 2^31.
    const int ibase = ((int)(b * H_ + 2u * ho) * W_ + (int)(2u * wo)) * C_
                      + (int)(4u * c4);
    const int obase = ((int)(b * HO_ + ho) * WO_ + (int)wo) * C_
                      + (int)(4u * c4);
    const int colS = C_;            // +1 in W:  +128 floats
    const int rowS = W_ * C_;       // +1 in H:  +28672 floats

    // Four 128-bit non-temporal loads (stream has zero reuse).
    v4f v00 = __builtin_nontemporal_load((const v4f*)(x + ibase));
    v4f v01 = __builtin_nontemporal_load((const v4f*)(x + ibase + colS));
    v4f v10 = __builtin_nontemporal_load((const v4f*)(x + ibase + rowS));
    v4f v11 = __builtin_nontemporal_load((const v4f*)(x + ibase + rowS + colS));

    const float K = 1.44269504088896340736f;        // log2(e)

    v4f r;
#pragma unroll
    for (int j = 0; j < 4; ++j) {
        float a0 = v00[j], a1 = v01[j], a2 = v10[j], a3 = v11[j];
        float m   = fmaxf(fmaxf(a0, a1), fmaxf(a2, a3));
        float nmK = -m * K;
        // exp(v - m) == exp2(K*v - K*m); hardware v_exp_f32.
        float e0 = __builtin_amdgcn_exp2f(__builtin_fmaf(a0, K, nmK));
        float e1 = __builtin_amdgcn_exp2f(__builtin_fmaf(a1, K, nmK));
        float e2 = __builtin_amdgcn_exp2f(__builtin_fmaf(a2, K, nmK));
        float e3 = __builtin_amdgcn_exp2f(__builtin_fmaf(a3, K, nmK));
        float den = (e0 + e1) + (e2 + e3);
        float num = __builtin_fmaf(a0, e0,
                    __builtin_fmaf(a1, e1,
                    __builtin_fmaf(a2, e2, a3 * e3)));
        r[j] = num * __builtin_amdgcn_rcpf(den);    // v_rcp_f32
    }

    __builtin_nontemporal_store(r, (v4f*)(out + obase));
}

extern "C" void kernel_launch(void* const* d_in, const int* in_sizes, int n_in,
                              void* d_out, int out_size, void* d_ws, size_t ws_size,
                              hipStream_t stream) {
    (void)in_sizes; (void)n_in; (void)out_size; (void)d_ws; (void)ws_size;
    const float* x = (const float*)d_in[0];
    float* out = (float*)d_out;

    dim3 block(256);                       // 8 waves = 8 output pixels per block
    dim3 grid(WO_ / 8, HO_, B_);           // 14 x 112 x 16 = 25,088 blocks, exact
    attn_pool2x2_kernel<<<grid, block, 0, stream>>>(x, out);
}